// NeuroODE_12429635355058
// MI455X (gfx1250) — compile-verified
//
#include <hip/hip_runtime.h>

// NeuroODE on MI455X (gfx1250, wave32).
//
// Reference: y_{k+1} = y_k + sub*(a*roll(y_k,1) - b*y_k + c*sin(t_k)), 8 substeps
// per output row, 2047 rows. Affine map M = d*I + e*S (S = circular shift):
//   M^8 = sum_j C(8,j) d^(8-j) e^j S^j   (9-tap one-sided circular conv)
//   forcing stays a uniform scalar since M*ones = (d+e)*ones.
// One outer step per 16-tile:  y' = P*y_tile + Q*y_tile_left + f
// with P[m][k] = c_{m-k}, Q[m][k] = c_{16+m-k}  -> 2 x (16x16x16) matmuls,
// done as 4+4 chained V_WMMA_F32_16X16X4_F32 per wave (16 tiles per wave,
// tiles are the 16 columns of the B operand). 8 waves = 2048 elements.
//
// v2 latency tuning (problem is a 2047-step dependence chain; HBM traffic is
// only 16 MB, irrelevant at 23.3 TB/s):
//  - double-buffered LDS state  -> ONE barrier per step instead of two
//  - global row-store issued AFTER the barrier -> the pre-barrier wait covers
//    only DScnt (LDS visibility); HBM store latency is hidden a full
//    iteration deep instead of sitting inside every barrier.

typedef float v2f __attribute__((ext_vector_type(2)));
typedef float v8f __attribute__((ext_vector_type(8)));

#define SAMPLE_NUM 2048
#define Y_NUM      2048
#define NTILES     128          // Y_NUM / 16
#define WAVES      8            // 16 tiles per wave
#define THREADS    (WAVES * 32)

// Select c[j] for runtime j in [0,8], else 0. Unrolled -> v_cndmask chain fed
// from SGPR taps; co-executes with the WMMA (XDL) pipe.
__device__ __forceinline__ float sel9(const float cc[9], int j) {
    float r = 0.0f;
#pragma unroll
    for (int i = 0; i < 9; ++i) r = (j == i) ? cc[i] : r;
    return r;
}

__global__ __launch_bounds__(THREADS)
void neuro_ode_wmma(const float* __restrict__ t,
                    const float* __restrict__ y0,
                    const float* __restrict__ w,
                    const float* __restrict__ rt,
                    float* __restrict__ out) {
    __shared__ float Y[2][Y_NUM];   // double-buffered state: 16 KB of 320 KB LDS

    const int tid  = threadIdx.x;
    const int wv   = tid >> 5;          // wave id 0..7
    const int ln   = tid & 31;          // lane  0..31
    const int col  = ln & 15;           // N index (tile within wave's group)
    const int half = ln >> 4;           // lane half 0/1
    const int m    = col;               // A-operand row index (lane % 16)

    const float a = w[0] * rt[0];
    const float b = w[1] * rt[1];
    const float c = w[2] * rt[2];

    // Row 0 = y0; also seed buffer 0.
    for (int i = tid; i < Y_NUM; i += THREADS) {
        float v = y0[i];
        Y[0][i] = v;
        out[i]  = v;
    }
    __syncthreads();

    const int tileC = (wv << 4) + col;                      // this lane's column tile
    const int tileL = (tileC + NTILES - 1) & (NTILES - 1);  // circular left neighbor
    const int rowIx = (wv << 8) + (col << 4) + (half << 3); // element index of acc[0]

    for (int n = 0; n < SAMPLE_NUM - 1; ++n) {
        const float tn  = t[n];
        const float tn1 = t[n + 1];
        const float sub = (tn1 - tn) * 0.125f;   // big_step / STEP_N
        const float d   = 1.0f - sub * b;
        const float e   = sub * a;

        // taps c_j = C(8,j) d^(8-j) e^j   (scalar-pipe work, loop-pipelined)
        float cc[9];
        {
            const float bin[9] = {1.f, 8.f, 28.f, 56.f, 70.f, 56.f, 28.f, 8.f, 1.f};
            float dp[9], ep[9];
            dp[0] = 1.0f; ep[0] = 1.0f;
#pragma unroll
            for (int j = 1; j < 9; ++j) { dp[j] = dp[j-1] * d; ep[j] = ep[j-1] * e; }
#pragma unroll
            for (int j = 0; j < 9; ++j) cc[j] = bin[j] * dp[8 - j] * ep[j];
        }

        // uniform forcing: f = sum_m s_m (d+e)^(7-m), Horner
        const float g = d + e;
        float f = 0.0f;
#pragma unroll
        for (int ms = 0; ms < 8; ++ms) {
            const float s = sub * c * __sinf(tn + sub * (float)ms);
            f = f * g + s;
        }

        const float* __restrict__ Yc = Y[n & 1];        // read buffer
        float*       __restrict__ Yn = Y[(n & 1) ^ 1];  // write buffer

        // Load B operands (old state).
        // B layout (4x16, K=4): lane L, vgpr v -> row k = v + 2*half, col N = L%16.
        v2f Byc[4], Byl[4];
#pragma unroll
        for (int q = 0; q < 4; ++q) {
            const int kk = (q << 2) + (half << 1);
            const float2 vc = *(const float2*)&Yc[(tileC << 4) + kk];
            const float2 vl = *(const float2*)&Yc[(tileL << 4) + kk];
            Byc[q][0] = vc.x; Byc[q][1] = vc.y;
            Byl[q][0] = vl.x; Byl[q][1] = vl.y;
        }

        // C = f everywhere (forcing), then accumulate P*Yc + Q*Yl.
        v8f acc;
#pragma unroll
        for (int i = 0; i < 8; ++i) acc[i] = f;

        // A layout (16x4, F32): lane L -> M = L%16; vgpr v -> K = v + 2*half.
#pragma unroll
        for (int q = 0; q < 4; ++q) {
            const int j0 = m - ((q << 2) + (half << 1));   // m - k(v=0)
            v2f Ap;
            Ap[0] = sel9(cc, j0);
            Ap[1] = sel9(cc, j0 - 1);
            acc = __builtin_amdgcn_wmma_f32_16x16x4_f32(
                false, Ap, false, Byc[q], (short)0, acc, false, false);
        }
#pragma unroll
        for (int q = 0; q < 4; ++q) {
            const int j0 = 16 + m - ((q << 2) + (half << 1));
            v2f Aq;
            Aq[0] = sel9(cc, j0);
            Aq[1] = sel9(cc, j0 - 1);
            acc = __builtin_amdgcn_wmma_f32_16x16x4_f32(
                false, Aq, false, Byl[q], (short)0, acc, false, false);
        }

        // D layout: lane L -> N = L%16; vgpr v -> M = v + 8*half.
        // Element = 256*wv + 16*N + M  ->  8 consecutive floats per lane.
        const float4 lo = make_float4(acc[0], acc[1], acc[2], acc[3]);
        const float4 hi = make_float4(acc[4], acc[5], acc[6], acc[7]);
        *(float4*)&Yn[rowIx]     = lo;
        *(float4*)&Yn[rowIx + 4] = hi;

        __syncthreads();  // publish new state (waits DS only; no HBM store yet)

        // HBM row store AFTER the barrier: drains in background, waited on
        // (storecnt) only at the next iteration's barrier — fully hidden.
        float* __restrict__ orow = out + (size_t)(n + 1) * Y_NUM;
        *(float4*)&orow[rowIx]     = lo;
        *(float4*)&orow[rowIx + 4] = hi;
    }
}

extern "C" void kernel_launch(void* const* d_in, const int* in_sizes, int n_in,
                              void* d_out, int out_size, void* d_ws, size_t ws_size,
                              hipStream_t stream) {
    (void)in_sizes; (void)n_in; (void)out_size; (void)d_ws; (void)ws_size;
    const float* t  = (const float*)d_in[0];
    const float* y0 = (const float*)d_in[1];
    const float* w  = (const float*)d_in[2];
    const float* r  = (const float*)d_in[3];
    float* out = (float*)d_out;

    // Single workgroup: the whole problem is one 2047-step dependence chain
    // over a 2048-float state; 8 waves cooperate through LDS + one barrier/step.
    neuro_ode_wmma<<<1, THREADS, 0, stream>>>(t, y0, w, r, out);
}